// EdgeToNodeAggregationLayer_59184649339043
// MI455X (gfx1250) — compile-verified
//
#include <hip/hip_runtime.h>
#include <stdint.h>

typedef float v2f __attribute__((ext_vector_type(2)));
typedef float v8f __attribute__((ext_vector_type(8)));

#define IN_CH      64
#define OUT_CH     64
#define LDS_STRIDE 66   // 64 + 2 pad floats: spreads the 64 LDS banks across rows

// ---------------------------------------------------------------------------
// Zero the accumulation buffer (harness poisons d_out with 0xAA).
// ---------------------------------------------------------------------------
__global__ void zero_out_kernel(float* __restrict__ out, int n) {
    int i = blockIdx.x * blockDim.x + threadIdx.x;
    if (i < n) out[i] = 0.0f;
}

// ---------------------------------------------------------------------------
// One wave32 per 16-edge tile:
//   proj[16,64] = A[16,64] x W^T[64,64]  via  V_WMMA_F32_16X16X4_F32
//   then per-element global_atomic_add_f32 scatter into node buckets.
//
// A-fragment (16x4 f32, ISA 7.12.2): lane m=lane&15, khalf=lane>>4,
//   VGPR0/1 = A[m][kk+2*khalf +0/+1]  -> contiguous float2 in memory (NT load;
//   256 MB stream must not evict the 12.8 MB output resident in L2).
// B-fragment (4x16 f32): lane n=lane&15 carries column o=nt*16+n,
//   VGPR0/1 = B[kk+2*khalf +0/+1][o] = W[o][kk+2*khalf +0/+1] -> float2 in LDS.
// C/D (16x16 f32): VGPR v holds row m=v+8*khalf, column n=lane&15.
//
// Scatter epilogue: node shuffles (8x ds_bpermute) and 32-bit byte offsets are
// hoisted out of the nt loop; full-tile validity is a wave-uniform branch so
// the hot path has zero per-atomic predication, and nt*16 folds into the
// atomic's immediate offset.
// ---------------------------------------------------------------------------
__global__ __launch_bounds__(256)
void edge_gemm_scatter_kernel(const float* __restrict__ ef,
                              const long long* __restrict__ row,
                              const float* __restrict__ W,
                              float* __restrict__ out,
                              int E, int num_tiles) {
    __shared__ float ldsW[OUT_CH * LDS_STRIDE];

    // Cooperative load of W (row-major [o][c], 16 KB) into padded LDS.
    for (int i = threadIdx.x; i < OUT_CH * IN_CH; i += blockDim.x) {
        int o = i >> 6, c = i & 63;
        ldsW[o * LDS_STRIDE + c] = W[i];
    }
    __syncthreads();

    const int lane  = threadIdx.x & 31;
    const int wave  = threadIdx.x >> 5;
    const int tile  = blockIdx.x * (blockDim.x >> 5) + wave;
    if (tile >= num_tiles) return;           // wave-uniform exit: EXEC stays all-1s

    const int m     = lane & 15;             // A-matrix row this lane feeds
    const int khalf = lane >> 4;             // which K-pair half
    const int base  = tile * 16;
    const int eg    = base + m;              // global edge id for row m
    const bool fulltile = (base + 16 <= E);  // wave-uniform

    // Node id for edge m; lanes 16..31 duplicate lanes 0..15.
    int mynode = 0;
    if (eg < E) mynode = (int)__builtin_nontemporal_load(&row[eg]);

    // ---- A fragments: 16 K-steps, each a float2, streamed non-temporally ----
    v2f a[16];
    const float* arow = ef + (size_t)eg * IN_CH + 2 * khalf;
    if (fulltile) {
        #pragma unroll
        for (int j = 0; j < 16; ++j)
            a[j] = __builtin_nontemporal_load((const v2f*)(arow + j * 4));
    } else {
        #pragma unroll
        for (int j = 0; j < 16; ++j) {
            if (eg < E) a[j] = __builtin_nontemporal_load((const v2f*)(arow + j * 4));
            else        { a[j].x = 0.0f; a[j].y = 0.0f; }
        }
    }

    const int n = lane & 15;                 // output column within N-tile

    // ---- hoisted scatter addressing: 8 shuffles + 8 unsigned offsets ----
    unsigned off[8];
    #pragma unroll
    for (int v = 0; v < 8; ++v) {
        int node = __shfl(mynode, v + 8 * khalf, 32);  // node id lives in lane mm
        off[v] = (unsigned)node * OUT_CH + (unsigned)n;
    }

    #pragma unroll
    for (int nt = 0; nt < 4; ++nt) {
        const float* brow = &ldsW[(nt * 16 + n) * LDS_STRIDE + 2 * khalf];
        v8f acc = {0.f, 0.f, 0.f, 0.f, 0.f, 0.f, 0.f, 0.f};
        #pragma unroll
        for (int j = 0; j < 16; ++j) {
            v2f b = *(const v2f*)(brow + j * 4);   // 8B-aligned ds_load_b64 pair
            acc = __builtin_amdgcn_wmma_f32_16x16x4_f32(
                /*neg_a=*/false, a[j], /*neg_b=*/false, b,
                /*c_mod=*/(short)0, acc, /*reuse_a=*/false, /*reuse_b=*/false);
        }
        // ---- scatter-add rows of the 16x16 C tile into node buckets ----
        if (fulltile) {
            #pragma unroll
            for (int v = 0; v < 8; ++v)
                atomicAdd(out + off[v] + nt * 16, acc[v]);   // nt*16 -> IOFFSET
        } else {
            #pragma unroll
            for (int v = 0; v < 8; ++v) {
                if (base + v + 8 * khalf < E)
                    atomicAdd(out + off[v] + nt * 16, acc[v]);
            }
        }
    }
}

// ---------------------------------------------------------------------------
// Inputs (setup_inputs order): 0=edge_features f32 [E*64], 1=edge_index i64
// [2*E] (row = first E), 2=num_nodes scalar, 3=W f32 [64*64].
// Output: node_features f32 [num_nodes*64].
// ---------------------------------------------------------------------------
extern "C" void kernel_launch(void* const* d_in, const int* in_sizes, int n_in,
                              void* d_out, int out_size, void* d_ws, size_t ws_size,
                              hipStream_t stream) {
    const float*     ef   = (const float*)d_in[0];
    const long long* eidx = (const long long*)d_in[1];   // int64 per reference
    const float*     W    = (const float*)d_in[3];
    float*           out  = (float*)d_out;

    const int E = in_sizes[0] / IN_CH;

    // 1) zero the output accumulator
    {
        int blocks = (out_size + 255) / 256;
        zero_out_kernel<<<blocks, 256, 0, stream>>>(out, out_size);
    }

    // 2) WMMA projection + atomic scatter. 8 waves / 256-thread block,
    //    one 16-edge tile per wave.
    {
        int tiles  = (E + 15) / 16;
        int wpb    = 256 / 32;
        int blocks = (tiles + wpb - 1) / wpb;
        edge_gemm_scatter_kernel<<<blocks, 256, 0, stream>>>(
            ef, eidx, W, out, E, tiles);
    }
}